// BatchGraphAttentionLayer_11295763988802
// MI455X (gfx1250) — compile-verified
//
#include <hip/hip_runtime.h>

typedef __attribute__((ext_vector_type(2))) float v2f;
typedef __attribute__((ext_vector_type(8))) float v8f;
typedef int v4i __attribute__((vector_size(16)));

#define IN_F  16384
#define OUT_F 256
#define NN    64
#define BB    32
#define ROWS  2048          // B*N
#define KSPLIT 8
#define KSLICE (IN_F / KSPLIT)   // 2048
#define KSTAGE 32
#define NST    (KSLICE / KSTAGE) // 64 stages per block
#define ALPHA 0.2f
#define NEGM  (-9.0e15f)

// Async global->LDS copy of 16 bytes per lane (GLOBAL_LOAD_ASYNC_TO_LDS_B128,
// tracked by ASYNCcnt). Builtin expects (v4i addrspace(1)*, v4i addrspace(3)*,
// imm offset, imm cpol). LDS pointer built from the low 32 bits of the flat
// address (flat LDS aperture: addr[31:0] == LDS offset).
__device__ __forceinline__ void async_copy_b128(const float* gsrc, float* ldst) {
  __builtin_amdgcn_global_load_async_to_lds_b128(
      (__attribute__((address_space(1))) v4i*)(unsigned long long)(uintptr_t)gsrc,
      (__attribute__((address_space(3))) v4i*)(unsigned int)(uintptr_t)ldst,
      0, 0);
}

// ---------------------------------------------------------------------------
// Kernel 1: partial Wh = h @ W over one K-slice of 2048. Split-K by 8:
// grid (32, 2, 8) = 512 blocks = 4096 waves.
// Block = 256 threads (8 waves). Block tile: 64(M) x 128(N). Wave tile: 16x64
// via four 16x16 f32 WMMA accumulators.
// Double-buffered LDS pipeline, K-stage = 32:
//   h tile staged via GLOBAL_LOAD_ASYNC_TO_LDS_B128 (no VGPR round-trip),
//   W tile staged via global->VGPR loads with the transpose ds_stores deferred
//   until AFTER the compute ds_loads (LDS queue is in-order per wave).
// ---------------------------------------------------------------------------
__global__ __launch_bounds__(256) void gat_gemm_hW(const float* __restrict__ h,
                                                   const float* __restrict__ W,
                                                   float* __restrict__ WhP) {
  __shared__ __align__(16) float sh [2][64][36];    // h tile  [buf][m][k]   18.0 KB
  __shared__ __align__(16) float swt[2][128][36];   // W^T tile [buf][n][k]  36.0 KB

  const int tid  = threadIdx.x;
  const int wave = tid >> 5;
  const int lane = tid & 31;
  const int l16  = lane & 15;
  const int hi   = lane >> 4;      // 0: lanes 0-15, 1: lanes 16-31
  const int m_sub = wave >> 1;     // 0..3  -> 16-row strip
  const int n_sub = wave & 1;      // 0..1  -> 64-col strip
  const long mbase = (long)blockIdx.x * 64;
  const int  nbase = blockIdx.y * 128;
  const int  kz    = blockIdx.z;   // K-slice index
  const int  kbeg  = kz * KSLICE;

  // staging coordinates
  const int hr = tid >> 3;            // 0..31 (h rows, 2 passes of 32)
  const int hc = (tid & 7) << 2;      // 0..28 (h k-cols, 16B chunks)
  const int wr = tid >> 5;            // 0..7  (W k-rows, 4 passes of 8)
  const int wc = (tid & 31) << 2;     // 0..124 (W n-cols)

  v8f acc0 = {}; v8f acc1 = {}; v8f acc2 = {}; v8f acc3 = {};
  float4 wreg[4];

  // issue async h copies for stage k0 into buffer buf
  auto stage_h = [&](int k0, int buf) {
    #pragma unroll
    for (int p = 0; p < 2; ++p)
      async_copy_b128(&h[(mbase + hr + p * 32) * (long)IN_F + k0 + hc],
                      &sh[buf][hr + p * 32][hc]);
  };
  // issue W global loads for stage k0 into registers
  auto fetch_w = [&](int k0) {
    #pragma unroll
    for (int p = 0; p < 4; ++p)
      wreg[p] = *(const float4*)&W[(long)(k0 + wr + p * 8) * OUT_F + nbase + wc];
  };
  // transpose-store W registers into buffer buf
  auto store_w = [&](int buf) {
    #pragma unroll
    for (int p = 0; p < 4; ++p) {
      const int kk = wr + p * 8;
      swt[buf][wc + 0][kk] = wreg[p].x;
      swt[buf][wc + 1][kk] = wreg[p].y;
      swt[buf][wc + 2][kk] = wreg[p].z;
      swt[buf][wc + 3][kk] = wreg[p].w;
    }
  };

  // ---- prologue: fully stage stage 0 into buffer 0 ----
  stage_h(kbeg, 0);
  fetch_w(kbeg);
  store_w(0);

  for (int i = 0; i < NST; ++i) {
    __builtin_amdgcn_s_wait_asynccnt(0);   // my async h copies done
    __syncthreads();                       // everyone's staging visible
    const int buf = i & 1;

    // kick off stage i+1 into the other buffer (overlaps with compute below)
    const bool more = (i + 1) < NST;
    if (more) {
      const int knext = kbeg + (i + 1) * KSTAGE;
      stage_h(knext, buf ^ 1);             // async, no VGPRs
      fetch_w(knext);                      // global loads in flight
    }

    // ---- compute on buffer buf ----
    const int arow  = m_sub * 16 + l16;
    const int cbase = n_sub * 64 + l16;
    #pragma unroll
    for (int kt = 0; kt < 8; ++kt) {
      const int kb = kt * 4 + hi * 2;
      const v2f a  = *(const v2f*)&sh [buf][arow       ][kb];
      const v2f b0 = *(const v2f*)&swt[buf][cbase +  0][kb];
      const v2f b1 = *(const v2f*)&swt[buf][cbase + 16][kb];
      const v2f b2 = *(const v2f*)&swt[buf][cbase + 32][kb];
      const v2f b3 = *(const v2f*)&swt[buf][cbase + 48][kb];
      acc0 = __builtin_amdgcn_wmma_f32_16x16x4_f32(false, a, false, b0, (short)0, acc0, false, false);
      acc1 = __builtin_amdgcn_wmma_f32_16x16x4_f32(false, a, false, b1, (short)0, acc1, false, false);
      acc2 = __builtin_amdgcn_wmma_f32_16x16x4_f32(false, a, false, b2, (short)0, acc2, false, false);
      acc3 = __builtin_amdgcn_wmma_f32_16x16x4_f32(false, a, false, b3, (short)0, acc3, false, false);
    }

    // deferred transpose-store of W for stage i+1 (after compute ds_loads so
    // the in-order LDS queue never blocks compute behind global latency)
    if (more) store_w(buf ^ 1);
  }

  // C/D layout: VGPR r -> M = r (lanes 0-15) / M = 8+r (lanes 16-31)
  float* Whz = WhP + (size_t)kz * ROWS * OUT_F;
  const long mrow = mbase + m_sub * 16 + hi * 8;
  const int  nc   = nbase + n_sub * 64 + l16;
  #pragma unroll
  for (int r = 0; r < 8; ++r) {
    Whz[(mrow + r) * OUT_F + nc +  0] = acc0[r];
    Whz[(mrow + r) * OUT_F + nc + 16] = acc1[r];
    Whz[(mrow + r) * OUT_F + nc + 32] = acc2[r];
    Whz[(mrow + r) * OUT_F + nc + 48] = acc3[r];
  }
}

// ---------------------------------------------------------------------------
// Kernel 2: deterministic split-K reduction (fixed summation order) fused with
// the row dots: Wh = sum_p WhP[p]; Wh1 = Wh @ a[:256]; Wh2 = Wh @ a[256:].
// One block (256 threads) per output row.
// ---------------------------------------------------------------------------
__global__ __launch_bounds__(256) void gat_reduce_rowdots(const float* __restrict__ WhP,
                                                          const float* __restrict__ a,
                                                          float* __restrict__ Wh,
                                                          float* __restrict__ Wh1,
                                                          float* __restrict__ Wh2) {
  __shared__ float red1[8];
  __shared__ float red2[8];
  const int row  = blockIdx.x;           // 0..2047
  const int c    = threadIdx.x;          // 0..255 (one column each)
  const int wave = c >> 5;
  const int lane = c & 31;

  const size_t ofs = (size_t)row * OUT_F + c;
  float v = 0.f;
  #pragma unroll
  for (int p = 0; p < KSPLIT; ++p)       // fixed order -> deterministic
    v += WhP[(size_t)p * ROWS * OUT_F + ofs];
  Wh[ofs] = v;

  float t1 = v * a[c];
  float t2 = v * a[OUT_F + c];
  #pragma unroll
  for (int off = 16; off > 0; off >>= 1) {
    t1 += __shfl_xor(t1, off, 32);
    t2 += __shfl_xor(t2, off, 32);
  }
  if (lane == 0) { red1[wave] = t1; red2[wave] = t2; }
  __syncthreads();
  if (c == 0) {
    float s1 = 0.f, s2 = 0.f;
    #pragma unroll
    for (int i = 0; i < 8; ++i) { s1 += red1[i]; s2 += red2[i]; }
    Wh1[row] = s1;
    Wh2[row] = s2;
  }
}

// ---------------------------------------------------------------------------
// Kernel 3: per batch b -> masked LeakyReLU logits, softmax over axis=1
// (column-wise), h_prime = att @ Wh[b] via f32 WMMA, ELU on store.
// One block per batch; att lives in LDS.
// ---------------------------------------------------------------------------
__global__ __launch_bounds__(256) void gat_attention(const float* __restrict__ Wh,
                                                     const float* __restrict__ Wh1,
                                                     const float* __restrict__ Wh2,
                                                     const int* __restrict__ adj,
                                                     float* __restrict__ out) {
  __shared__ float att[NN][68];
  __shared__ float s1[NN];
  __shared__ float s2[NN];
  const int b   = blockIdx.x;
  const int tid = threadIdx.x;

  if (tid < NN) {
    s1[tid] = Wh1[b * NN + tid];
    s2[tid] = Wh2[b * NN + tid];
  }
  __syncthreads();

  if (tid < NN) {                         // thread j owns column j (softmax axis=1)
    const int j = tid;
    float mx = -3.0e38f;
    for (int i = 0; i < NN; ++i) {
      float e = s1[i] + s2[j];
      e = e > 0.f ? e : ALPHA * e;        // LeakyReLU
      const float x = (adj[i * NN + j] > 0) ? e : NEGM;
      att[i][j] = x;
      mx = fmaxf(mx, x);
    }
    float sum = 0.f;
    for (int i = 0; i < NN; ++i) {
      const float p = __expf(att[i][j] - mx);
      att[i][j] = p;
      sum += p;
    }
    const float inv = 1.0f / sum;
    for (int i = 0; i < NN; ++i) att[i][j] *= inv;
  }
  __syncthreads();

  // h_prime = att(64x64) @ Wh[b](64x256) with V_WMMA_F32_16X16X4_F32.
  const int wave  = tid >> 5;
  const int lane  = tid & 31;
  const int l16   = lane & 15;
  const int hi    = lane >> 4;
  const int m_sub = wave & 3;             // 4 row strips of 16
  const int nhalf = wave >> 2;            // 2 col halves of 128
  const float* whb = Wh + (long)b * NN * OUT_F;
  float* ob = out + (long)b * NN * OUT_F;

  for (int nt = 0; nt < 8; ++nt) {
    const int ncol = nhalf * 128 + nt * 16 + l16;
    v8f acc = {};
    #pragma unroll
    for (int kt = 0; kt < 16; ++kt) {
      const int kb = kt * 4 + hi * 2;
      const v2f a = *(const v2f*)&att[m_sub * 16 + l16][kb];
      v2f bf;
      bf.x = whb[(kb + 0) * OUT_F + ncol];
      bf.y = whb[(kb + 1) * OUT_F + ncol];
      acc = __builtin_amdgcn_wmma_f32_16x16x4_f32(false, a, false, bf, (short)0, acc, false, false);
    }
    const int mrow = m_sub * 16 + hi * 8;
    #pragma unroll
    for (int r = 0; r < 8; ++r) {
      const float x = acc[r];
      ob[(long)(mrow + r) * OUT_F + ncol] = x > 0.f ? x : (__expf(x) - 1.0f);   // ELU
    }
  }
}

// ---------------------------------------------------------------------------
extern "C" void kernel_launch(void* const* d_in, const int* in_sizes, int n_in,
                              void* d_out, int out_size, void* d_ws, size_t ws_size,
                              hipStream_t stream) {
  const float* h   = (const float*)d_in[0];   // [32,64,16384] f32
  const int*   adj = (const int*)  d_in[1];   // [64,64] i32
  const float* W   = (const float*)d_in[2];   // [16384,256] f32
  const float* a   = (const float*)d_in[3];   // [512,1] f32
  float* out = (float*)d_out;                 // [32,64,256] f32

  // Workspace layout (floats):
  //   WhP : KSPLIT * ROWS * OUT_F   (split-K partials, ~16.8 MB)
  //   Wh  : ROWS * OUT_F            (reduced projection, 2 MB)
  //   Wh1 : ROWS
  //   Wh2 : ROWS
  float* WhP = (float*)d_ws;
  float* Wh  = WhP + (size_t)KSPLIT * ROWS * OUT_F;
  float* Wh1 = Wh  + (size_t)ROWS * OUT_F;
  float* Wh2 = Wh1 + ROWS;

  dim3 g1(ROWS / 64, OUT_F / 128, KSPLIT);    // (32, 2, 8) = 512 blocks
  gat_gemm_hW       <<<g1,   256, 0, stream>>>(h, W, WhP);
  gat_reduce_rowdots<<<ROWS, 256, 0, stream>>>(WhP, a, Wh, Wh1, Wh2);
  gat_attention     <<<BB,   256, 0, stream>>>(Wh, Wh1, Wh2, adj, out);
}